// MolVisGNN_45552423141675
// MI455X (gfx1250) — compile-verified
//
#include <hip/hip_runtime.h>
#include <hip/hip_bf16.h>

typedef float v2f __attribute__((ext_vector_type(2)));
typedef float v8f __attribute__((ext_vector_type(8)));

// ---------------------------------------------------------------------------
// CDNA5 f32 WMMA: D(16x16) = A(16x4) * B(4x16) + C   (wave32, exact f32)
// A frag: lane l holds M=l&15, K = j + 2*(l>>4)  (j = component 0/1)
// B frag: lane l holds N=l&15, K = j + 2*(l>>4)
// C/D:    vgpr r, lane l: M = r + 8*(l>>4), N = l&15
// ---------------------------------------------------------------------------
__device__ __forceinline__ v8f wmma_f32(v2f a, v2f b, v8f c) {
#if __has_builtin(__builtin_amdgcn_wmma_f32_16x16x4_f32)
  return __builtin_amdgcn_wmma_f32_16x16x4_f32(false, a, false, b, (short)0, c,
                                               false, false);
#else
  return c; // host pass only
#endif
}

// ---------------------------------------------------------------------------
// CDNA5 async global->LDS copy (GLOBAL_LOAD_ASYNC_TO_LDS_B32, ASYNCcnt).
// Probe round 2 showed the builtin exists with param0 = AS(1) int*.
// ---------------------------------------------------------------------------
#if defined(__AMDGCN__) &&                                          \
    __has_builtin(__builtin_amdgcn_global_load_async_to_lds_b32) && \
    __has_builtin(__builtin_amdgcn_s_wait_asynccnt)
#define HAVE_ASYNC_LDS 1
typedef __attribute__((address_space(1))) int* gas_ptr_t;
typedef __attribute__((address_space(3))) int* las_ptr_t;
#else
#define HAVE_ASYNC_LDS 0
#endif

__device__ __forceinline__ void async_copy_b32(const float* gsrc, float* ldst) {
#if HAVE_ASYNC_LDS
  __builtin_amdgcn_global_load_async_to_lds_b32(
      (gas_ptr_t)(void*)(float*)gsrc, (las_ptr_t)(void*)ldst, 0, 0);
#else
  *ldst = *gsrc;
#endif
}

__device__ __forceinline__ void async_copy_wait() {
#if HAVE_ASYNC_LDS
  __builtin_amdgcn_s_wait_asynccnt(0);
#endif
}

#define NN 4096

// ===========================================================================
// Kernel 1: conv3d(SAME,3x3x3, 128->64) on one 2x4x4 slice + ReLU + running
// mean into e3.  Implicit GEMM: M=32 positions, N=64 out-ch, K=128*27.
// Block = 256 threads = 8 waves, one sample per wave.  Per-channel weight
// tile (28x64, tap-padded) is DOUBLE-BUFFERED in LDS and staged with async
// global->LDS copies overlapped with the WMMA work of the current channel.
// Activations live one-per-lane; im2col A fragments come from ds_bpermute.
// ===========================================================================
__global__ __launch_bounds__(256) void conv3d_e3_kernel(
    const float* __restrict__ drug3d,  // (4096,128,6,4,4)
    const float* __restrict__ w_fr, const float* __restrict__ b_fr,
    const float* __restrict__ w_bb, const float* __restrict__ b_bb,
    const float* __restrict__ w_tl, const float* __restrict__ b_tl,
    float* __restrict__ e3)            // (4096,64) pre-zeroed
{
  __shared__ float Bl[2][28 * 64];  // Bl[buf][o*28 + t]; t==27 stays 0
  __shared__ float bias_s[64];

  const int slice = blockIdx.y;
  const int tid   = threadIdx.x;
  const int wave  = tid >> 5;
  const int lane  = tid & 31;
  const int half  = lane >> 4;
  const int l15   = lane & 15;
  const int n     = blockIdx.x * 8 + wave;

  const float* w_s = (slice == 0) ? w_fr : (slice == 1) ? w_bb : w_tl;
  const float* b_s = (slice == 0) ? b_fr : (slice == 1) ? b_bb : b_tl;
  if (tid < 64) bias_s[tid] = b_s[tid];
  // zero the tap-padding column (t == 27) of both buffers, once
  for (int i = tid; i < 128; i += 256) {
    int bf = i >> 6, o = i & 63;
    Bl[bf][o * 28 + 27] = 0.0f;
  }

  v8f acc[2][4];
  {
    v8f z = {};
    for (int m = 0; m < 2; ++m)
      for (int t = 0; t < 4; ++t) acc[m][t] = z;
  }

  const float* xbase = drug3d + (size_t)n * 12288 + slice * 32;

  // stage channel 0 into buffer 0
  for (int i = tid; i < 64 * 27; i += 256) {
    int o = i / 27, t = i - o * 27;
    async_copy_b32(w_s + (size_t)o * 3456 + t, &Bl[0][o * 28 + t]);
  }

  for (int c = 0; c < 128; ++c) {
    const int cur = c & 1;
    async_copy_wait();   // this wave's staged tile is in LDS
    __syncthreads();     // all waves' staging + previous compute complete

    if (c + 1 < 128) {   // prefetch next channel's weights into other buffer
      const float* wsrc = w_s + (c + 1) * 27;
      float* dbuf = &Bl[cur ^ 1][0];
      for (int i = tid; i < 64 * 27; i += 256) {
        int o = i / 27, t = i - o * 27;
        async_copy_b32(wsrc + (size_t)o * 3456 + t, &dbuf[o * 28 + t]);
      }
    }

    // channel activations: lane = output-position index p = d*16+h*4+w
    float xc = xbase[c * 96 + lane];

#pragma unroll
    for (int ks = 0; ks < 7; ++ks) {
      v2f afr[2];
#pragma unroll
      for (int m = 0; m < 2; ++m) {
#pragma unroll
        for (int j = 0; j < 2; ++j) {
          int t  = ks * 4 + 2 * half + j;     // tap index, 0..27
          int tc = (t < 27) ? t : 26;         // clamp for safe decomposition
          int p  = m * 16 + l15;              // output position
          int od = p >> 4, oh = (p >> 2) & 3, ow = p & 3;
          int kd = tc / 9, r9 = tc - kd * 9;
          int kh = r9 / 3, kw = r9 - kh * 3;
          int id = od + kd - 1, ih = oh + kh - 1, iw = ow + kw - 1;
          int sp = id * 16 + ih * 4 + iw;
          float g = __shfl(xc, sp);           // unconditional: EXEC all ones
          bool ok = (t < 27) & (id >= 0) & (id < 2) & (ih >= 0) & (ih < 4) &
                    (iw >= 0) & (iw < 4);
          afr[m][j] = ok ? g : 0.0f;
        }
      }
#pragma unroll
      for (int nt = 0; nt < 4; ++nt) {
        v2f bfr;
#pragma unroll
        for (int j = 0; j < 2; ++j) {
          int o  = nt * 16 + l15;
          int kk = ks * 4 + 2 * half + j;     // 0..27 (27 is zero-padded)
          bfr[j] = Bl[cur][o * 28 + kk];
        }
        acc[0][nt] = wmma_f32(afr[0], bfr, acc[0][nt]);
        acc[1][nt] = wmma_f32(afr[1], bfr, acc[1][nt]);
      }
    }
  }

  // bias + ReLU, sum over the 32 positions, accumulate mean into e3
#pragma unroll
  for (int nt = 0; nt < 4; ++nt) {
    int o   = nt * 16 + l15;
    float b = bias_s[o];
    float s = 0.0f;
#pragma unroll
    for (int m = 0; m < 2; ++m)
#pragma unroll
      for (int r = 0; r < 8; ++r) {
        float v = acc[m][nt][r] + b;
        s += (v > 0.0f) ? v : 0.0f;
      }
    s += __shfl_xor(s, 16);
    if (lane < 16) atomicAdd(&e3[n * 64 + o], s * (1.0f / 96.0f));
  }
}

// ===========================================================================
// Kernel 2: adaptive-avg-pool the three branches to length 256
// ===========================================================================
__global__ __launch_bounds__(256) void pool_branches_kernel(
    const float* __restrict__ xd, const float* __restrict__ d2d,
    const float* __restrict__ e3, float* __restrict__ d1o,
    float* __restrict__ d2o, float* __restrict__ e3o)
{
  int n = blockIdx.x, j = threadIdx.x;
  const float* x = xd + (size_t)n * 1024;
  d1o[n * 256 + j] = 0.25f * (x[4*j] + x[4*j+1] + x[4*j+2] + x[4*j+3]);
  const float* y = d2d + (size_t)n * 512;
  d2o[n * 256 + j] = 0.5f * (y[2*j] + y[2*j+1]);
  e3o[n * 256 + j] = e3[n * 64 + (j >> 2)];
}

// ===========================================================================
// Kernel 3: gate MLP (256->128->64->32, mean, sigmoid) per branch, softmax
// over branches, gated concat (768) pooled to drug128_pre (groups of 6).
// ===========================================================================
__global__ __launch_bounds__(256) void gates_kernel(
    const float* __restrict__ b1, const float* __restrict__ b2,
    const float* __restrict__ b3, const float* __restrict__ g1w,
    const float* __restrict__ g1b, const float* __restrict__ g2w,
    const float* __restrict__ g2b, const float* __restrict__ g3w,
    const float* __restrict__ g3b, float* __restrict__ drug128)
{
  __shared__ float v[256], h1[128], h2[64], h3[32];
  __shared__ float logit[3];
  int n = blockIdx.x, tid = threadIdx.x;

  for (int b = 0; b < 3; ++b) {
    __syncthreads();
    const float* src = (b == 0) ? b1 : (b == 1) ? b2 : b3;
    v[tid] = src[(size_t)n * 256 + tid];
    __syncthreads();
    if (tid < 128) {
      float s = g1b[tid];
      const float* w = g1w + tid * 256;
      for (int k = 0; k < 256; ++k) s += v[k] * w[k];
      h1[tid] = s;
    }
    __syncthreads();
    if (tid < 64) {
      float s = g2b[tid];
      const float* w = g2w + tid * 128;
      for (int k = 0; k < 128; ++k) s += h1[k] * w[k];
      h2[tid] = s;
    }
    __syncthreads();
    if (tid < 32) {
      float s = g3b[tid];
      const float* w = g3w + tid * 64;
      for (int k = 0; k < 64; ++k) s += h2[k] * w[k];
      h3[tid] = s;
    }
    __syncthreads();
    if (tid == 0) {
      float s = 0.0f;
      for (int k = 0; k < 32; ++k) s += h3[k];
      s *= (1.0f / 32.0f);
      logit[b] = 1.0f / (1.0f + __expf(-s));
    }
  }
  __syncthreads();

  float l0 = logit[0], l1 = logit[1], l2 = logit[2];
  float mx = fmaxf(l0, fmaxf(l1, l2));
  float e0 = __expf(l0 - mx), e1 = __expf(l1 - mx), e2 = __expf(l2 - mx);
  float inv = 1.0f / (e0 + e1 + e2);
  float gw0 = e0 * inv, gw1 = e1 * inv, gw2 = e2 * inv;

  if (tid < 128) {
    float s = 0.0f;
    for (int k = 0; k < 6; ++k) {
      int i = tid * 6 + k;
      int b = i >> 8, e = i & 255;
      float ev = (b == 0) ? b1[(size_t)n * 256 + e]
               : (b == 1) ? b2[(size_t)n * 256 + e]
                          : b3[(size_t)n * 256 + e];
      float g = (b == 0) ? gw0 : (b == 1) ? gw1 : gw2;
      s += ev * g;
    }
    drug128[n * 128 + tid] = s * (1.0f / 6.0f);
  }
}

// ===========================================================================
// Kernel 4: batchnorm statistics (population mean/var over N=4096)
// ===========================================================================
__global__ __launch_bounds__(128) void bn_stats_kernel(
    const float* __restrict__ x, float* __restrict__ sum,
    float* __restrict__ sumsq)
{
  int f = threadIdx.x;
  float s = 0.0f, q = 0.0f;
  for (int n = blockIdx.x; n < NN; n += gridDim.x) {
    float v = x[n * 128 + f];
    s += v; q += v * v;
  }
  atomicAdd(&sum[f], s);
  atomicAdd(&sumsq[f], q);
}

// ===========================================================================
// Kernel 5: apply BN, run convt1d/relu/maxpool4 tower, residual, upsample x2
// ===========================================================================
__global__ __launch_bounds__(128) void tower_kernel(
    const float* __restrict__ pre, const float* __restrict__ sum,
    const float* __restrict__ sumsq, const float* __restrict__ bng,
    const float* __restrict__ bnb, const float* __restrict__ w1,
    const float* __restrict__ c1b, const float* __restrict__ w2,
    const float* __restrict__ c2b, const float* __restrict__ w3,
    const float* __restrict__ c3b, float* __restrict__ drug256)
{
  __shared__ float xs[128];
  __shared__ float A[32 * 127];   // big ping buffer
  __shared__ float B[32 * 31];    // pong buffer
  int n = blockIdx.x, tid = threadIdx.x;

  {
    float m   = sum[tid] * (1.0f / 4096.0f);
    float var = sumsq[tid] * (1.0f / 4096.0f) - m * m;
    float a   = bng[tid] * rsqrtf(var + 1e-5f);
    xs[tid]   = (pre[(size_t)n * 128 + tid] - m) * a + bnb[tid];
  }
  __syncthreads();

  // stage 1: (1,128) -> (32,127), relu
  for (int i = tid; i < 32 * 127; i += 128) {
    int o = i / 127, t = i - o * 127;
    float v = xs[t] * w1[o * 2 + 1] + xs[t + 1] * w1[o * 2] + c1b[o];
    A[i] = (v > 0.0f) ? v : 0.0f;
  }
  __syncthreads();
  // maxpool4 -> (32,31)
  for (int i = tid; i < 32 * 31; i += 128) {
    int o = i / 31, u = i - o * 31;
    const float* p = &A[o * 127 + u * 4];
    B[i] = fmaxf(fmaxf(p[0], p[1]), fmaxf(p[2], p[3]));
  }
  __syncthreads();
  // stage 2: (32,31) -> (64,30), relu
  for (int i = tid; i < 64 * 30; i += 128) {
    int o = i / 30, t = i - o * 30;
    float s = c2b[o];
    for (int c = 0; c < 32; ++c)
      s += B[c * 31 + t] * w2[c * 128 + o * 2 + 1] +
           B[c * 31 + t + 1] * w2[c * 128 + o * 2];
    A[i] = (s > 0.0f) ? s : 0.0f;
  }
  __syncthreads();
  // maxpool4 -> (64,7)
  for (int i = tid; i < 64 * 7; i += 128) {
    int o = i / 7, u = i - o * 7;
    const float* p = &A[o * 30 + u * 4];
    B[i] = fmaxf(fmaxf(p[0], p[1]), fmaxf(p[2], p[3]));
  }
  __syncthreads();
  // stage 3: (64,7) -> (128,4 of 6), relu (maxpool4 only keeps t=0..3)
  for (int i = tid; i < 128 * 4; i += 128) {
    int o = i / 4, t = i - o * 4;
    float s = c3b[o];
    for (int c = 0; c < 64; ++c)
      s += B[c * 7 + t] * w3[c * 256 + o * 2 + 1] +
           B[c * 7 + t + 1] * w3[c * 256 + o * 2];
    A[i] = (s > 0.0f) ? s : 0.0f;
  }
  __syncthreads();
  {
    int o = tid;
    float h = fmaxf(fmaxf(A[o * 4], A[o * 4 + 1]),
                    fmaxf(A[o * 4 + 2], A[o * 4 + 3]));
    float dn = h + xs[o];
    drug256[(size_t)n * 256 + 2 * o]     = dn;   // aap1d 128 -> 256
    drug256[(size_t)n * 256 + 2 * o + 1] = dn;
  }
}

// ===========================================================================
// Kernel 6: generic WMMA-f32 GEMM  out = act(A(MxK) @ W(NxK)^T + bias + add)
// one 16x64 tile per wave, 4 waves / block, grid = (M/64, ceil(N/64)).
// K-loop split into an unchecked main body and a single checked tail.
// ===========================================================================
__global__ __launch_bounds__(128) void gemm_wmma_kernel(
    const float* __restrict__ A, const float* __restrict__ W,
    const float* __restrict__ bias, const float* __restrict__ addsrc,
    float* __restrict__ out, int M, int N, int K, int lrelu)
{
  int wave = threadIdx.x >> 5, lane = threadIdx.x & 31;
  int l15 = lane & 15, half = lane >> 4;
  int mbase = (blockIdx.x * 4 + wave) * 16;
  int nbase = blockIdx.y * 64;
  if (mbase >= M) return;

  v8f acc[4];
  {
    v8f z = {};
    for (int t = 0; t < 4; ++t) acc[t] = z;
  }
  int ntc = (N - nbase + 15) >> 4;
  ntc = (ntc > 4) ? 4 : ntc;

  const float* arow = A + (size_t)(mbase + l15) * K + 2 * half;
  const int kmain = K & ~3;
  for (int kk = 0; kk < kmain; kk += 4) {
    __builtin_prefetch(arow + kk + 64, 0, 0);
    v2f af;
    af[0] = arow[kk];
    af[1] = arow[kk + 1];
    for (int nt = 0; nt < ntc; ++nt) {
      const float* wrow = W + (size_t)(nbase + nt * 16 + l15) * K + 2 * half;
      v2f bf;
      bf[0] = wrow[kk];
      bf[1] = wrow[kk + 1];
      acc[nt] = wmma_f32(af, bf, acc[nt]);
    }
  }
  if (kmain < K) {  // tail (K % 4 != 0), bounds-checked
    v2f af;
#pragma unroll
    for (int j = 0; j < 2; ++j) {
      int k = kmain + 2 * half + j;
      af[j] = (k < K) ? A[(size_t)(mbase + l15) * K + k] : 0.0f;
    }
    for (int nt = 0; nt < ntc; ++nt) {
      v2f bf;
#pragma unroll
      for (int j = 0; j < 2; ++j) {
        int k = kmain + 2 * half + j;
        bf[j] = (k < K) ? W[(size_t)(nbase + nt * 16 + l15) * K + k] : 0.0f;
      }
      acc[nt] = wmma_f32(af, bf, acc[nt]);
    }
  }

  for (int nt = 0; nt < ntc; ++nt) {
    int col = nbase + nt * 16 + l15;
    float bv = bias ? bias[col] : 0.0f;
#pragma unroll
    for (int r = 0; r < 8; ++r) {
      int row = mbase + r + 8 * half;
      float v = acc[nt][r] + bv;
      if (addsrc) v += addsrc[(size_t)row * N + col];
      if (lrelu) v = (v > 0.0f) ? v : 0.2f * v;
      out[(size_t)row * N + col] = v;
    }
  }
}

// ===========================================================================
// SAGE aggregation: segment-sum of source features + edge counts + normalize
// ===========================================================================
__global__ __launch_bounds__(256) void seg_sum_kernel(
    const float* __restrict__ feat, const int* __restrict__ edge,
    float* __restrict__ agg, int E, int F4)
{
  long long id = (long long)blockIdx.x * 256 + threadIdx.x;
  long long total = (long long)E * F4;
  if (id >= total) return;
  int e = (int)(id / F4);
  int r = (int)(id - (long long)e * F4);
  int s = edge[e], d = edge[E + e];
  const float* src = feat + (size_t)s * (F4 * 4) + r * 4;
  float* dst = agg + (size_t)d * (F4 * 4) + r * 4;
  atomicAdd(&dst[0], src[0]);
  atomicAdd(&dst[1], src[1]);
  atomicAdd(&dst[2], src[2]);
  atomicAdd(&dst[3], src[3]);
}

__global__ __launch_bounds__(256) void seg_cnt_kernel(
    const int* __restrict__ edge, float* __restrict__ cnt, int E)
{
  int e = blockIdx.x * 256 + threadIdx.x;
  if (e < E) atomicAdd(&cnt[edge[E + e]], 1.0f);
}

__global__ __launch_bounds__(256) void normalize_kernel(
    const float* __restrict__ agg, const float* __restrict__ cnt,
    float* __restrict__ aggn, int F)
{
  int id = blockIdx.x * 256 + threadIdx.x;
  if (id >= NN * F) return;
  int row = id / F;
  float c = cnt[row];
  c = (c < 1.0f) ? 1.0f : c;
  aggn[id] = agg[id] / c;
}

// ===========================================================================
// Host orchestration
// ===========================================================================
extern "C" void kernel_launch(void* const* d_in, const int* in_sizes, int n_in,
                              void* d_out, int out_size, void* d_ws,
                              size_t ws_size, hipStream_t stream)
{
  const float* x_drug   = (const float*)d_in[0];
  const float* x_mirna  = (const float*)d_in[1];
  const float* drug_2d  = (const float*)d_in[2];
  const float* drug_3d  = (const float*)d_in[3];
  const float* sp_fr_w  = (const float*)d_in[4];
  const float* sp_fr_b  = (const float*)d_in[5];
  const float* sp_bb_w  = (const float*)d_in[6];
  const float* sp_bb_b  = (const float*)d_in[7];
  const float* sp_tl_w  = (const float*)d_in[8];
  const float* sp_tl_b  = (const float*)d_in[9];
  const float* g1_w = (const float*)d_in[10];
  const float* g1_b = (const float*)d_in[11];
  const float* g2_w = (const float*)d_in[12];
  const float* g2_b = (const float*)d_in[13];
  const float* g3_w = (const float*)d_in[14];
  const float* g3_b = (const float*)d_in[15];
  const float* bn_g = (const float*)d_in[16];
  const float* bn_b = (const float*)d_in[17];
  const float* ct1_w = (const float*)d_in[18];
  const float* ct1_b = (const float*)d_in[19];
  const float* ct2_w = (const float*)d_in[20];
  const float* ct2_b = (const float*)d_in[21];
  const float* ct3_w = (const float*)d_in[22];
  const float* ct3_b = (const float*)d_in[23];
  const float* lne_w = (const float*)d_in[24];
  const float* lne_b = (const float*)d_in[25];
  const float* c_lw[6] = {(const float*)d_in[26], (const float*)d_in[29],
                          (const float*)d_in[32], (const float*)d_in[35],
                          (const float*)d_in[38], (const float*)d_in[41]};
  const float* c_lb[6] = {(const float*)d_in[27], (const float*)d_in[30],
                          (const float*)d_in[33], (const float*)d_in[36],
                          (const float*)d_in[39], (const float*)d_in[42]};
  const float* c_rw[6] = {(const float*)d_in[28], (const float*)d_in[31],
                          (const float*)d_in[34], (const float*)d_in[37],
                          (const float*)d_in[40], (const float*)d_in[43]};
  const int* edge_md = (const int*)d_in[44];
  const int* edge_dm = (const int*)d_in[45];
  const int E = in_sizes[44] / 2;

  float* out = (float*)d_out;

  // workspace layout (floats)
  float* ws = (float*)d_ws;
  size_t off = 0;
  auto alloc = [&](size_t nflt) { float* p = ws + off; off += nflt; return p; };
  float* e3      = alloc((size_t)NN * 64);
  float* d1p     = alloc((size_t)NN * 256);
  float* d2p     = alloc((size_t)NN * 256);
  float* e3u     = alloc((size_t)NN * 256);
  float* drug128 = alloc((size_t)NN * 128);
  float* bnsum   = alloc(128);
  float* bnsq    = alloc(128);
  float* drug256 = alloc((size_t)NN * 256);
  float* mirna256= alloc((size_t)NN * 256);
  float* agg     = alloc((size_t)NN * 256);
  float* cnt     = alloc(NN);
  float* aggn    = alloc((size_t)NN * 256);
  float* tmp     = alloc((size_t)NN * 128);
  float* d1b     = alloc((size_t)NN * 128);
  float* m1b     = alloc((size_t)NN * 128);
  float* d2b     = alloc((size_t)NN * 64);
  float* m2b     = alloc((size_t)NN * 64);
  (void)off; (void)ws_size; (void)n_in; (void)out_size;

  // ---- 3D conv branch -> e3 (4096,64)
  (void)hipMemsetAsync(e3, 0, (size_t)NN * 64 * sizeof(float), stream);
  {
    dim3 grid(NN / 8, 3);
    conv3d_e3_kernel<<<grid, 256, 0, stream>>>(drug_3d, sp_fr_w, sp_fr_b,
                                               sp_bb_w, sp_bb_b, sp_tl_w,
                                               sp_tl_b, e3);
  }
  // ---- branch pooling to 256
  pool_branches_kernel<<<NN, 256, 0, stream>>>(x_drug, drug_2d, e3, d1p, d2p,
                                               e3u);
  // ---- gates + gated concat pooled to 128
  gates_kernel<<<NN, 256, 0, stream>>>(d1p, d2p, e3u, g1_w, g1_b, g2_w, g2_b,
                                       g3_w, g3_b, drug128);
  // ---- batchnorm stats
  (void)hipMemsetAsync(bnsum, 0, 128 * sizeof(float), stream);
  (void)hipMemsetAsync(bnsq, 0, 128 * sizeof(float), stream);
  bn_stats_kernel<<<64, 128, 0, stream>>>(drug128, bnsum, bnsq);
  // ---- BN apply + convt tower + residual + upsample -> drug256
  tower_kernel<<<NN, 128, 0, stream>>>(drug128, bnsum, bnsq, bn_g, bn_b,
                                       ct1_w, ct1_b, ct2_w, ct2_b, ct3_w,
                                       ct3_b, drug256);
  // ---- mirna embedding: (4096,382) @ (256,382)^T + b -> mirna256
  {
    dim3 grid(NN / 64, 4);
    gemm_wmma_kernel<<<grid, 128, 0, stream>>>(x_mirna, lne_w, lne_b, nullptr,
                                               mirna256, NN, 256, 382, 0);
  }

  // ---- SAGE layers
  auto sage = [&](const float* xsrc, const float* xdst, const int* edge,
                  const float* lw, const float* lb, const float* rw, int ci,
                  int co, float* outp) {
    (void)hipMemsetAsync(agg, 0, (size_t)NN * ci * sizeof(float), stream);
    (void)hipMemsetAsync(cnt, 0, NN * sizeof(float), stream);
    int F4 = ci / 4;
    long long tot = (long long)E * F4;
    seg_sum_kernel<<<(unsigned)((tot + 255) / 256), 256, 0, stream>>>(
        xsrc, edge, agg, E, F4);
    seg_cnt_kernel<<<(E + 255) / 256, 256, 0, stream>>>(edge, cnt, E);
    normalize_kernel<<<(NN * ci + 255) / 256, 256, 0, stream>>>(agg, cnt,
                                                                aggn, ci);
    dim3 grid(NN / 64, (co + 63) / 64);
    gemm_wmma_kernel<<<grid, 128, 0, stream>>>(aggn, lw, lb, nullptr, tmp, NN,
                                               co, ci, 0);
    gemm_wmma_kernel<<<grid, 128, 0, stream>>>(xdst, rw, nullptr, tmp, outp,
                                               NN, co, ci, 1);
  };

  // layer 1: 256 -> 128
  sage(mirna256, drug256, edge_md, c_lw[0], c_lb[0], c_rw[0], 256, 128, d1b);
  sage(drug256, mirna256, edge_dm, c_lw[1], c_lb[1], c_rw[1], 256, 128, m1b);
  // layer 2: 128 -> 64
  sage(m1b, d1b, edge_md, c_lw[2], c_lb[2], c_rw[2], 128, 64, d2b);
  sage(d1b, m1b, edge_dm, c_lw[3], c_lb[3], c_rw[3], 128, 64, m2b);
  // layer 3: 64 -> 32, outputs straight into d_out (d then m)
  sage(m2b, d2b, edge_md, c_lw[4], c_lb[4], c_rw[4], 64, 32, out);
  sage(d2b, m2b, edge_dm, c_lw[5], c_lb[5], c_rw[5], 64, 32,
       out + (size_t)NN * 32);
}